// MambaLM_52999896433020
// MI455X (gfx1250) — compile-verified
//
#include <hip/hip_runtime.h>
#include <hip/hip_bf16.h>
#include <math.h>

typedef __attribute__((ext_vector_type(2))) float v2f;
typedef __attribute__((ext_vector_type(8))) float v8f;

#define NLAYERS 4
#define DMODEL  768
#define DINNER  1536
#define DSTATE  16
#define DTRANK  48
#define DCONV   4
#define BB      2
#define LL      512
#define MROWS   (BB * LL)      /* 1024 */
#define VOCABN  32000
#define SSMW    (DTRANK + 2 * DSTATE)  /* 80 */

// ---------------------------------------------------------------------------
// Embedding gather: h[m, :] = embed[ids[m], :]
// ---------------------------------------------------------------------------
__global__ void embed_gather_k(const int* __restrict__ ids,
                               const float* __restrict__ embed,
                               float* __restrict__ h) {
    int m = blockIdx.x;
    long id = ids[m];
    const float* src = embed + id * (long)DMODEL;
    float* dst = h + (long)m * DMODEL;
    for (int i = threadIdx.x; i < DMODEL; i += blockDim.x) dst[i] = src[i];
}

// ---------------------------------------------------------------------------
// RMSNorm: one block (256 thr = 8 wave32) per row of 768
// ---------------------------------------------------------------------------
__global__ __launch_bounds__(256) void rmsnorm_k(const float* __restrict__ x,
                                                 const float* __restrict__ w,
                                                 float* __restrict__ o) {
    __shared__ float red[8];
    __shared__ float sInv;
    int m = blockIdx.x;
    const float* xr = x + (long)m * DMODEL;
    float s = 0.f;
    for (int i = threadIdx.x; i < DMODEL; i += blockDim.x) { float v = xr[i]; s += v * v; }
    for (int off = 16; off > 0; off >>= 1) s += __shfl_down(s, off, 32);
    if ((threadIdx.x & 31) == 0) red[threadIdx.x >> 5] = s;
    __syncthreads();
    if (threadIdx.x == 0) {
        float tot = 0.f;
        for (int i = 0; i < 8; ++i) tot += red[i];
        sInv = rsqrtf(tot / (float)DMODEL + 1e-5f);
    }
    __syncthreads();
    float inv = sInv;
    float* orow = o + (long)m * DMODEL;
    for (int i = threadIdx.x; i < DMODEL; i += blockDim.x) orow[i] = xr[i] * inv * w[i];
}

// ---------------------------------------------------------------------------
// Generic f32 WMMA GEMM: C[M,N] (+)= A[M,K] * B[N,K]^T
// K contiguous in both operands. One wave -> 16(M) x 16*NSUB(N) strip.
// NSUB is compile-time: the inner loop has NO control flow (EXEC stays
// all-ones per WMMA requirement) and accumulators stay pinned in registers.
// Single A and B row pointers (subtile offsets are hoisted constants) so
// address-space inference yields global_load_b64 for all operands.
// A fragment (16x4 f32, ISA 7.12.2): lane m: K={k0,k0+1}; lane m+16: K={k0+2,k0+3}
// B fragment: column n of B (= row n of [N,K]) striped identically.
// C/D layout: VGPR r -> row r + 8*(lane>=16), col = lane&15.
// ---------------------------------------------------------------------------
enum { EPI_STORE = 0, EPI_ADDC = 1, EPI_BIAS_SOFTPLUS = 2 };

template <int EPI, int NSUB>
__global__ __launch_bounds__(256) void gemm_wmma_f32_k(
    const float* __restrict__ A, int lda,
    const float* __restrict__ B, int ldb,
    float* __restrict__ C, int ldc,
    const float* __restrict__ bias,
    int M, int N, int K) {
    const int lane  = threadIdx.x & 31;
    const int gwave = blockIdx.x * (blockDim.x >> 5) + (threadIdx.x >> 5);
    const int ntn   = N / (16 * NSUB);          // N divides exactly (launch invariant)
    const int mt    = gwave / ntn;
    if (mt * 16 >= M) return;                   // wave-uniform guard
    const int n0   = (gwave % ntn) * (16 * NSUB);
    const int m0   = mt << 4;
    const int row  = lane & 15;
    const int kh   = (lane >> 4) << 1;          // 0 or 2

    const float* Ap = A + (long)(m0 + row) * lda + kh;
    const float* Bp = B + (long)(n0 + row) * ldb + kh;
    const long bstride = (long)16 * ldb;        // floats between N-subtiles

    v8f acc[NSUB] = {};

#pragma unroll 2
    for (int k0 = 0; k0 < K; k0 += 4) {
        v2f a;
        a.x = Ap[0];
        a.y = Ap[1];
        Ap += 4;
#pragma unroll
        for (int j = 0; j < NSUB; ++j) {
            const float* p = Bp + (long)j * bstride;
            v2f bv;
            bv.x = p[0];
            bv.y = p[1];
            acc[j] = __builtin_amdgcn_wmma_f32_16x16x4_f32(
                false, a, false, bv, (short)0, acc[j], false, false);
        }
        Bp += 4;
    }

    const int ccol  = lane & 15;
    const int rbase = (lane >> 4) << 3;          // 0 or 8
#pragma unroll
    for (int j = 0; j < NSUB; ++j) {
        const int n = n0 + j * 16 + ccol;
        const float bn = (EPI == EPI_BIAS_SOFTPLUS) ? bias[n] : 0.f;
#pragma unroll
        for (int r = 0; r < 8; ++r) {
            long cidx = (long)(m0 + rbase + r) * ldc + n;
            float v = acc[j][r];
            if (EPI == EPI_ADDC) v += C[cidx];
            if (EPI == EPI_BIAS_SOFTPLUS) {
                v += bn;
                v = (v > 20.f) ? v : log1pf(expf(v));   // softplus
            }
            C[cidx] = v;
        }
    }
}

// ---------------------------------------------------------------------------
// Causal depthwise conv (k=4) + bias + SiLU over proj[:, :, :DINNER]
// ---------------------------------------------------------------------------
__global__ void conv_silu_k(const float* __restrict__ proj,
                            const float* __restrict__ cw,
                            const float* __restrict__ cb,
                            float* __restrict__ hact) {
    long idx = (long)blockIdx.x * blockDim.x + threadIdx.x;
    if (idx >= (long)MROWS * DINNER) return;
    int ch   = (int)(idx % DINNER);
    int rowi = (int)(idx / DINNER);       // b*L + t
    int t    = rowi & (LL - 1);
    float s = cb[ch];
#pragma unroll
    for (int j = 0; j < DCONV; ++j) {
        int tt = t - (DCONV - 1) + j;
        if (tt >= 0)
            s += cw[ch * DCONV + j] * proj[(long)(rowi - (DCONV - 1) + j) * (2 * DINNER) + ch];
    }
    hact[idx] = s / (1.f + expf(-s));      // silu
}

// ---------------------------------------------------------------------------
// Selective scan: one thread per (b, channel); 16 states in registers,
// B_t / C_t staged in LDS per timestep. Fuses D-skip + SiLU gate.
// ---------------------------------------------------------------------------
__global__ __launch_bounds__(256) void scan_k(
    const float* __restrict__ dtsp, const float* __restrict__ hact,
    const float* __restrict__ ssm,  const float* __restrict__ proj,
    const float* __restrict__ A_log, const float* __restrict__ Dsk,
    float* __restrict__ y) {
    __shared__ float sBC[2 * DSTATE];
    const int b  = blockIdx.x / (DINNER / 256);
    const int ch = (blockIdx.x % (DINNER / 256)) * 256 + threadIdx.x;

    float Av[DSTATE], st[DSTATE];
#pragma unroll
    for (int s = 0; s < DSTATE; ++s) {
        Av[s] = -expf(A_log[(long)ch * DSTATE + s]);
        st[s] = 0.f;
    }
    const float dskip = Dsk[ch];

    for (int t = 0; t < LL; ++t) {
        const int rowi = b * LL + t;
        if (threadIdx.x < 2 * DSTATE)
            sBC[threadIdx.x] = ssm[(long)rowi * SSMW + DTRANK + threadIdx.x];
        __syncthreads();

        const float dt   = dtsp[(long)rowi * DINNER + ch];
        const float hv   = hact[(long)rowi * DINNER + ch];
        const float dthv = dt * hv;
        float acc = 0.f;
#pragma unroll
        for (int s = 0; s < DSTATE; ++s) {
            st[s] = expf(dt * Av[s]) * st[s] + dthv * sBC[s];
            acc  += st[s] * sBC[DSTATE + s];
        }
        const float g = proj[(long)rowi * (2 * DINNER) + DINNER + ch];
        y[(long)rowi * DINNER + ch] = (acc + dskip * hv) * (g / (1.f + expf(-g)));
        __syncthreads();
    }
}

// ---------------------------------------------------------------------------
// Host-side launcher helpers
// ---------------------------------------------------------------------------
static void run_gemm(int epi, const float* A, int lda, const float* B, int ldb,
                     float* C, int ldc, const float* bias,
                     int M, int N, int K, hipStream_t stream) {
    const int nsub   = (N % 64 == 0) ? 4 : 1;   // N always a multiple of 16 here
    const int waves  = (M / 16) * (N / (16 * nsub));
    const int blocks = (waves + 7) / 8;         // 8 wave32 per 256-thread block
    dim3 g(blocks), blk(256);
    if (nsub == 4) {
        if (epi == EPI_STORE)
            gemm_wmma_f32_k<EPI_STORE, 4><<<g, blk, 0, stream>>>(A, lda, B, ldb, C, ldc, bias, M, N, K);
        else if (epi == EPI_ADDC)
            gemm_wmma_f32_k<EPI_ADDC, 4><<<g, blk, 0, stream>>>(A, lda, B, ldb, C, ldc, bias, M, N, K);
        else
            gemm_wmma_f32_k<EPI_BIAS_SOFTPLUS, 4><<<g, blk, 0, stream>>>(A, lda, B, ldb, C, ldc, bias, M, N, K);
    } else {
        if (epi == EPI_STORE)
            gemm_wmma_f32_k<EPI_STORE, 1><<<g, blk, 0, stream>>>(A, lda, B, ldb, C, ldc, bias, M, N, K);
        else if (epi == EPI_ADDC)
            gemm_wmma_f32_k<EPI_ADDC, 1><<<g, blk, 0, stream>>>(A, lda, B, ldb, C, ldc, bias, M, N, K);
        else
            gemm_wmma_f32_k<EPI_BIAS_SOFTPLUS, 1><<<g, blk, 0, stream>>>(A, lda, B, ldb, C, ldc, bias, M, N, K);
    }
}

extern "C" void kernel_launch(void* const* d_in, const int* in_sizes, int n_in,
                              void* d_out, int out_size, void* d_ws, size_t ws_size,
                              hipStream_t stream) {
    const int*   ids       = (const int*)d_in[0];
    const float* embed     = (const float*)d_in[1];
    const float* norm_f_w  = (const float*)d_in[2];
    const float* in_proj_w = (const float*)d_in[3];
    const float* conv_w    = (const float*)d_in[4];
    const float* conv_b    = (const float*)d_in[5];
    const float* x_proj_w  = (const float*)d_in[6];
    const float* dt_proj_w = (const float*)d_in[7];
    const float* dt_proj_b = (const float*)d_in[8];
    const float* A_log     = (const float*)d_in[9];
    const float* D_skip    = (const float*)d_in[10];
    const float* out_proj_w= (const float*)d_in[11];
    const float* norm_w    = (const float*)d_in[12];
    float* logits = (float*)d_out;

    // scratch carve-out (floats)
    float* ws   = (float*)d_ws;
    float* h    = ws;                                   // [1024, 768]  residual
    float* xn   = h    + (long)MROWS * DMODEL;          // [1024, 768]  normed
    float* proj = xn   + (long)MROWS * DMODEL;          // [1024, 3072]
    float* hact = proj + (long)MROWS * 2 * DINNER;      // [1024, 1536]
    float* ssm  = hact + (long)MROWS * DINNER;          // [1024, 80]
    float* dtsp = ssm  + (long)MROWS * SSMW;            // [1024, 1536]
    float* ybuf = dtsp + (long)MROWS * DINNER;          // [1024, 1536]

    embed_gather_k<<<dim3(MROWS), dim3(256), 0, stream>>>(ids, embed, h);

    for (int i = 0; i < NLAYERS; ++i) {
        rmsnorm_k<<<dim3(MROWS), dim3(256), 0, stream>>>(h, norm_w + (long)i * DMODEL, xn);

        // proj = xn @ in_proj_w^T : [1024, 3072]
        run_gemm(EPI_STORE, xn, DMODEL,
                 in_proj_w + (long)i * 2 * DINNER * DMODEL, DMODEL,
                 proj, 2 * DINNER, nullptr, MROWS, 2 * DINNER, DMODEL, stream);

        conv_silu_k<<<dim3((MROWS * DINNER + 255) / 256), dim3(256), 0, stream>>>(
            proj, conv_w + (long)i * DINNER * DCONV, conv_b + (long)i * DINNER, hact);

        // ssm = hact @ x_proj_w^T : [1024, 80]  (N=80 -> NSUB=1 path)
        run_gemm(EPI_STORE, hact, DINNER,
                 x_proj_w + (long)i * SSMW * DINNER, DINNER,
                 ssm, SSMW, nullptr, MROWS, SSMW, DINNER, stream);

        // dtsp = softplus(ssm[:, :48] @ dt_proj_w^T + b) : [1024, 1536]
        run_gemm(EPI_BIAS_SOFTPLUS, ssm, SSMW,
                 dt_proj_w + (long)i * DINNER * DTRANK, DTRANK,
                 dtsp, DINNER, dt_proj_b + (long)i * DINNER,
                 MROWS, DINNER, DTRANK, stream);

        scan_k<<<dim3(BB * (DINNER / 256)), dim3(256), 0, stream>>>(
            dtsp, hact, ssm, proj,
            A_log + (long)i * DINNER * DSTATE, D_skip + (long)i * DINNER, ybuf);

        // h += ybuf @ out_proj_w^T : [1024, 768]
        run_gemm(EPI_ADDC, ybuf, DINNER,
                 out_proj_w + (long)i * DMODEL * DINNER, DINNER,
                 h, DMODEL, nullptr, MROWS, DMODEL, DINNER, stream);
    }

    rmsnorm_k<<<dim3(MROWS), dim3(256), 0, stream>>>(h, norm_f_w, xn);

    // logits = xn @ embed^T : [1024, 32000]
    run_gemm(EPI_STORE, xn, DMODEL, embed, DMODEL,
             logits, VOCABN, nullptr, MROWS, VOCABN, DMODEL, stream);
}